// GEARSNetwork_11098195493604
// MI455X (gfx1250) — compile-verified
//
#include <hip/hip_runtime.h>

// GEARS forward for MI455X (gfx1250, wave32). fp32 WMMA (V_WMMA_F32_16X16X4_F32)
// for all dense linears; BN fused into GEMM prologue/epilogue; tile-redundancy
// of the reference exploited so only the recovery MLP runs at N=640000 scale.

#define HD    128
#define GG    5000
#define PP    5000
#define BATCH 128
#define NTOT  (BATCH*GG)   // 640000
#define NE1   100000
#define NE2   100000
#define EPS   1e-5f
#define WPB   4            // waves per block in WMMA GEMM kernels

typedef __attribute__((ext_vector_type(2))) float v2f;
typedef __attribute__((ext_vector_type(8))) float v8f;

// ---------------------------------------------------------------- utilities
__global__ void renorm_rows(const float* __restrict__ X, float* __restrict__ Y, int R) {
  int wid  = (blockIdx.x * blockDim.x + threadIdx.x) >> 5;
  int lane = threadIdx.x & 31;
  if (wid >= R) return;
  const float* xr = X + (size_t)wid * HD;
  float s = 0.f;
  for (int h = lane; h < HD; h += 32) { float v = xr[h]; s += v * v; }
  for (int o = 16; o; o >>= 1) s += __shfl_down(s, o);
  s = __shfl(s, 0);
  float sc = fminf(1.f, 1.f / (sqrtf(s) + 1e-7f));
  float* yr = Y + (size_t)wid * HD;
  for (int h = lane; h < HD; h += 32) yr[h] = xr[h] * sc;
}

__global__ void colstats(const float* __restrict__ X, int R, int C, float w,
                         float* __restrict__ sum, float* __restrict__ sq) {
  int i = blockIdx.x * blockDim.x + threadIdx.x;
  if (i >= R * C) return;
  int c = i % C;
  float v = X[i];
  atomicAdd(&sum[c], w * v);
  atomicAdd(&sq[c], w * v * v);
}

__global__ void bn_finalize(const float* __restrict__ sum, const float* __restrict__ sq,
                            float count, const float* __restrict__ gamma,
                            const float* __restrict__ beta,
                            float* __restrict__ scale, float* __restrict__ shift, int C) {
  int c = blockIdx.x * blockDim.x + threadIdx.x;
  if (c >= C) return;
  float m = sum[c] / count;
  float v = sq[c] / count - m * m;
  float s = gamma[c] * rsqrtf(v + EPS);
  scale[c] = s;
  shift[c] = beta[c] - m * s;
}

__global__ void bn_apply(const float* __restrict__ X, const float* __restrict__ scale,
                         const float* __restrict__ shift, float* __restrict__ Y,
                         int R, int C, int relu) {
  int i = blockIdx.x * blockDim.x + threadIdx.x;
  if (i >= R * C) return;
  int c = i % C;
  float v = scale[c] * X[i] + shift[c];
  Y[i] = relu ? fmaxf(v, 0.f) : v;
}

__global__ void axpy02(const float* __restrict__ A, const float* __restrict__ Bp,
                       float* __restrict__ Y, int n) {
  int i = blockIdx.x * blockDim.x + threadIdx.x;
  if (i < n) Y[i] = A[i] + 0.2f * Bp[i];
}

__global__ void gather_rows(const float* __restrict__ src, const int* __restrict__ idx,
                            float* __restrict__ dst, int R) {
  int i = blockIdx.x * blockDim.x + threadIdx.x;
  if (i >= R * HD) return;
  int r = i >> 7, h = i & 127;
  dst[i] = src[(size_t)idx[r] * HD + h];
}

// ------------------------------------------------------------- graph pieces
__global__ void deg_init(float* deg, int n) {
  int i = blockIdx.x * blockDim.x + threadIdx.x;
  if (i < n) deg[i] = 1.f;           // self-loop weight
}
__global__ void deg_acc(const int* __restrict__ col, const float* __restrict__ w,
                        float* __restrict__ deg, int E) {
  int e = blockIdx.x * blockDim.x + threadIdx.x;
  if (e < E) atomicAdd(&deg[col[e]], w[e]);
}
__global__ void edge_norm(const int* __restrict__ row, const int* __restrict__ col,
                          const float* __restrict__ w, const float* __restrict__ deg,
                          float* __restrict__ nrm, int E) {
  int e = blockIdx.x * blockDim.x + threadIdx.x;
  if (e < E) nrm[e] = rsqrtf(deg[row[e]]) * w[e] * rsqrtf(deg[col[e]]);
}
__global__ void agg_init(const float* __restrict__ x, const float* __restrict__ deg,
                         float* __restrict__ agg, int R) {
  int i = blockIdx.x * blockDim.x + threadIdx.x;
  if (i >= R * HD) return;
  agg[i] = x[i] / deg[i >> 7];       // self loop: dinv^2 = 1/deg
}
__global__ void agg_edges(const int* __restrict__ row, const int* __restrict__ col,
                          const float* __restrict__ nrm, const float* __restrict__ x,
                          float* __restrict__ agg, int E) {
  int i = blockIdx.x * blockDim.x + threadIdx.x;
  if (i >= E * HD) return;
  int e = i >> 7, h = i & 127;
  atomicAdd(&agg[(size_t)col[e] * HD + h], x[(size_t)row[e] * HD + h] * nrm[e]);
}

// ------------------------------------------------------ generic WMMA GEMM
// C[M,N] = act(A[M,K]) @ W(ldw)[N,K]^T + bias ; optional BN-in + ReLU-in on A,
// optional ReLU-out, optional weighted column stats (for training-mode BN).
template <int K, bool BNIN, bool RELUIN, bool RELUOUT, bool STATS>
__global__ __launch_bounds__(32 * WPB)
void gemm_wmma(const float* __restrict__ A, const float* __restrict__ W, int ldw,
               const float* __restrict__ bias, float* __restrict__ C, int M, int Nout,
               const float* __restrict__ isc, const float* __restrict__ ish,
               float* __restrict__ osum, float* __restrict__ osq, float statw) {
  int wave = threadIdx.x >> 5;
  int tile = blockIdx.x * WPB + wave;
  int ntm = (M + 15) >> 4, ntn = (Nout + 15) >> 4;
  if (tile >= ntm * ntn) return;
  int tm = tile / ntn, tn = tile - tm * ntn;
  int lane = threadIdx.x & 31, half = lane >> 4, l16 = lane & 15;
  int arow = tm * 16 + l16;  if (arow >= M) arow = M - 1;
  int bcol = tn * 16 + l16;  bool colok = (tn * 16 + l16) < Nout;
  int bcolc = colok ? bcol : (Nout - 1);
  const float* Ar = A + (size_t)arow * K;
  const float* Wr = W + (size_t)bcolc * ldw;
  v8f acc = {};
  for (int kk = 0; kk < K; kk += 4) {
    int ka = kk + 2 * half;
    float a0 = Ar[ka], a1 = Ar[ka + 1];
    if (BNIN)   { a0 = a0 * isc[ka] + ish[ka]; a1 = a1 * isc[ka + 1] + ish[ka + 1]; }
    if (RELUIN) { a0 = fmaxf(a0, 0.f); a1 = fmaxf(a1, 0.f); }
    v2f av; av.x = a0; av.y = a1;
    v2f bv; bv.x = Wr[ka]; bv.y = Wr[ka + 1];
    acc = __builtin_amdgcn_wmma_f32_16x16x4_f32(false, av, false, bv, (short)0, acc,
                                                false, false);
  }
  float bb = bias ? bias[bcolc] : 0.f;
  float ls = 0.f, lq = 0.f;
  for (int r = 0; r < 8; ++r) {
    int row = tm * 16 + r + 8 * half;
    float v = acc[r] + bb;
    if (RELUOUT) v = fmaxf(v, 0.f);
    if (row < M && colok) {
      C[(size_t)row * Nout + bcol] = v;
      if (STATS) { ls += v; lq += v * v; }
    }
  }
  if (STATS && colok) {
    atomicAdd(&osum[bcol], statw * ls);
    atomicAdd(&osq[bcol], statw * lq);
  }
}

// ------------------------------- recovery layer-1 GEMM with synthesized A
// A row n=(b,g): relu(scale_pb[k]*((b==0?SA:SB)[g,k]+F[b,k]) + shift_pb[k])
__global__ __launch_bounds__(32 * WPB)
void recov_l1_wmma(const float* __restrict__ SA, const float* __restrict__ SB,
                   const float* __restrict__ F, const float* __restrict__ sc,
                   const float* __restrict__ sh, const float* __restrict__ W,
                   const float* __restrict__ bias, float* __restrict__ C,
                   float* __restrict__ osum, float* __restrict__ osq) {
  const int ntn = 256 / 16, ntm = NTOT / 16;
  int wave = threadIdx.x >> 5;
  int tile = blockIdx.x * WPB + wave;
  if (tile >= ntm * ntn) return;
  int tm = tile / ntn, tn = tile - tm * ntn;
  int lane = threadIdx.x & 31, half = lane >> 4, l16 = lane & 15;
  int arow = tm * 16 + l16;
  int b = arow / GG, g = arow - b * GG;
  const float* S  = (b == 0 ? SA : SB) + (size_t)g * HD;
  const float* Fr = F + (size_t)b * HD;
  int bcol = tn * 16 + l16;
  const float* Wr = W + (size_t)bcol * HD;
  v8f acc = {};
  for (int kk = 0; kk < HD; kk += 4) {
    int ka = kk + 2 * half;
    v2f av, bv;
    av.x = fmaxf(sc[ka] * (S[ka] + Fr[ka]) + sh[ka], 0.f);
    av.y = fmaxf(sc[ka + 1] * (S[ka + 1] + Fr[ka + 1]) + sh[ka + 1], 0.f);
    bv.x = Wr[ka]; bv.y = Wr[ka + 1];
    acc = __builtin_amdgcn_wmma_f32_16x16x4_f32(false, av, false, bv, (short)0, acc,
                                                false, false);
  }
  float bb = bias[bcol], ls = 0.f, lq = 0.f;
  for (int r = 0; r < 8; ++r) {
    int row = tm * 16 + r + 8 * half;
    float v = acc[r] + bb;
    C[(size_t)row * 256 + bcol] = v;
    ls += v; lq += v * v;
  }
  atomicAdd(&osum[bcol], ls);
  atomicAdd(&osq[bcol], lq);
}

// closed-form bn_pb stats from set/fused column sums (+ cross term)
__global__ void pb_stats(const float* sA, const float* qA, const float* sB,
                         const float* qB, const float* sF, const float* qF,
                         const float* fused, const float* g_pb, const float* b_pb,
                         float* scale, float* shift) {
  int h = blockIdx.x * blockDim.x + threadIdx.x;
  if (h >= HD) return;
  float f0 = fused[h];
  float sum = sA[h] + 127.f * sB[h] + (float)GG * sF[h];
  float sq  = qA[h] + 127.f * qB[h] + (float)GG * qF[h]
            + 2.f * (f0 * sA[h] + (sF[h] - f0) * sB[h]);
  float cnt = (float)NTOT;
  float m = sum / cnt, v = sq / cnt - m * m;
  float s = g_pb[h] * rsqrtf(v + EPS);
  scale[h] = s; shift[h] = b_pb[h] - m * s;
}

// per-node decode: w[b,g] = sum_h bn(Oraw[n,h]) * iw1[g,h] + ib1[g]
__global__ void decode1(const float* __restrict__ O, const float* __restrict__ sc,
                        const float* __restrict__ sh, const float* __restrict__ iw1,
                        const float* __restrict__ ib1, float* __restrict__ wmat) {
  int wid  = (blockIdx.x * blockDim.x + threadIdx.x) >> 5;
  int lane = threadIdx.x & 31;
  if (wid >= NTOT) return;
  int g = wid % GG;
  const float* o = O + (size_t)wid * HD;
  const float* w = iw1 + (size_t)g * HD;
  float s = 0.f;
  for (int h = lane; h < HD; h += 32) s += (sc[h] * o[h] + sh[h]) * w[h];
  for (int o2 = 16; o2; o2 >>= 1) s += __shfl_down(s, o2);
  if (lane == 0) wmat[wid] = s + ib1[g];
}

// cross L1: cg1[b,o] = sum_g wmat[b,g]*cw1[o,g] + cb1[o]  (K = G = 5000)
__global__ void cross_l1(const float* __restrict__ wmat, const float* __restrict__ cw1,
                         const float* __restrict__ cb1, float* __restrict__ cg1) {
  int wid  = (blockIdx.x * blockDim.x + threadIdx.x) >> 5;
  int lane = threadIdx.x & 31;
  if (wid >= BATCH * HD) return;
  int b = wid / HD, o = wid - b * HD;
  const float* wr = wmat + (size_t)b * GG;
  const float* cr = cw1 + (size_t)o * GG;
  float s = 0.f;
  for (int g = lane; g < GG; g += 32) s += wr[g] * cr[g];
  for (int o2 = 16; o2; o2 >>= 1) s += __shfl_down(s, o2);
  if (lane == 0) cg1[wid] = s + cb1[o];
}

__global__ void small_linear(const float* __restrict__ X, const float* __restrict__ W,
                             const float* __restrict__ bias, float* __restrict__ Y,
                             int R, int K, int O) {
  int i = blockIdx.x * blockDim.x + threadIdx.x;
  if (i >= R * O) return;
  int r = i / O, o = i - r * O;
  const float* xr = X + (size_t)r * K;
  const float* wr = W + (size_t)o * K;
  float s = bias[o];
  for (int k = 0; k < K; ++k) s += xr[k] * wr[k];
  Y[i] = s;
}

__global__ void final_out(const float* __restrict__ T, const float* __restrict__ wmat,
                          const float* __restrict__ iw2, const float* __restrict__ xin,
                          float* __restrict__ out, int n) {
  int i = blockIdx.x * blockDim.x + threadIdx.x;
  if (i >= n) return;
  int g = i % GG;
  out[i] = T[i] + wmat[i] * iw2[(size_t)g * 129] + xin[i];
}

// --------------------------------------------------------------------- host
static inline int gblk(int M, int Nout) {
  int tiles = ((M + 15) >> 4) * ((Nout + 15) >> 4);
  return (tiles + WPB - 1) / WPB;
}
static inline int cdiv(long n, int b) { return (int)((n + b - 1) / b); }

extern "C" void kernel_launch(void* const* d_in, const int* in_sizes, int n_in,
                              void* d_out, int out_size, void* d_ws, size_t ws_size,
                              hipStream_t stream) {
  (void)in_sizes; (void)n_in; (void)out_size; (void)ws_size;
  const float* x      = (const float*)d_in[0];
  const int*   p_idx  = (const int*)d_in[1];
  const int*   e_co   = (const int*)d_in[2];   // [2,E1] rows then cols
  const float* w_co   = (const float*)d_in[3];
  const int*   e_go   = (const int*)d_in[4];
  const float* w_go   = (const float*)d_in[5];
  int i = 6;
  auto F = [&](void) { return (const float*)d_in[i++]; };
  const float* gene_emb = F(); const float* pert_emb = F(); const float* pos_emb = F();
  const float* coexW = F();  const float* coexB = F();
  const float* goW   = F();  const float* goB   = F();
  const float* bnEg = F(); const float* bnEb = F();
  const float* bnPg = F(); const float* bnPb = F();
  const float *e_w1=F(),*e_b1=F(),*e_g1=F(),*e_bb1=F(),*e_w2=F(),*e_b2=F(),*e_g2=F(),*e_bb2=F();
  const float *pf_w1=F(),*pf_b1=F(),*pf_g1=F(),*pf_bb1=F(),*pf_w2=F(),*pf_b2=F(),*pf_g2=F(),*pf_bb2=F();
  const float *r_w1=F(),*r_b1=F(),*r_g1=F(),*r_bb1=F(),*r_w2=F(),*r_b2=F(),*r_g2=F(),*r_bb2=F();
  const float *c_w1=F(),*c_b1=F(),*c_g1=F(),*c_bb1=F(),*c_w2=F(),*c_b2=F(),*c_g2=F(),*c_bb2=F();
  const float* iw1 = F(); const float* ib1 = F();
  const float* iw2 = F(); const float* ib2 = F();
  float* out = (float*)d_out;

  float* ws = (float*)d_ws;
  size_t off = 0;
  auto A = [&](size_t n) { float* p = ws + off; off += n; return p; };
  const size_t GH = (size_t)GG * HD, PH = (size_t)PP * HD, BH = (size_t)BATCH * HD;
  float* rg    = A(GH); float* rp   = A(GH); float* rpe  = A(PH);
  float* gnorm = A(GH);
  float* posA  = A(GH); float* posA2 = A(GH); float* posB = A(GH); float* posB2 = A(GH);
  float* tmpG  = A(GH);
  float* pge1  = A(PH); float* pge2 = A(PH); float* tmpP = A(PH);
  float* degG  = A(GG); float* nrm1 = A(NE1);
  float* degP  = A(PP); float* nrm2 = A(NE2);
  float* inA   = A(GH); float* inB  = A(GH);
  float* hA    = A(GH); float* hB   = A(GH);
  float* outA  = A(GH); float* outB = A(GH);
  float* fIn   = A(BH); float* fh   = A(BH); float* fused = A(BH);
  float* wmat  = A((size_t)NTOT);
  float* cg1   = A(BH); float* cgh  = A(BH); float* cg = A(BH);
  float* Tbuf  = A((size_t)NTOT);
  float* ST    = A(48 * 256);                      // BN stat slots
  float* H1    = A((size_t)NTOT * 256);            // 655 MB
  float* Oraw  = A((size_t)NTOT * HD);             // 327 MB
  auto S = [&](int k) { return ST + (size_t)k * 256; };

  hipMemsetAsync(ST, 0, 48 * 256 * sizeof(float), stream);

  // --- renorm embeddings
  renorm_rows<<<cdiv((long)GG * 32, 256), 256, 0, stream>>>(gene_emb, rg, GG);
  renorm_rows<<<cdiv((long)GG * 32, 256), 256, 0, stream>>>(pos_emb, rp, GG);
  renorm_rows<<<cdiv((long)PP * 32, 256), 256, 0, stream>>>(pert_emb, rpe, PP);

  // --- bn_emb + relu on gene rows (tiling leaves stats unchanged)
  colstats<<<cdiv(GH, 256), 256, 0, stream>>>(rg, GG, HD, 1.f, S(0), S(1));
  bn_finalize<<<1, 128, 0, stream>>>(S(0), S(1), (float)GG, bnEg, bnEb, S(2), S(3), HD);
  bn_apply<<<cdiv(GH, 256), 256, 0, stream>>>(rg, S(2), S(3), gnorm, GG, HD, 1);

  // --- co-expression graph normalization (shared by both layers)
  deg_init<<<cdiv(GG, 256), 256, 0, stream>>>(degG, GG);
  deg_acc<<<cdiv(NE1, 256), 256, 0, stream>>>(e_co + NE1, w_co, degG, NE1);
  edge_norm<<<cdiv(NE1, 256), 256, 0, stream>>>(e_co, e_co + NE1, w_co, degG, nrm1, NE1);
  // layer 0 (copy 0 = real SGConv, copies>=1 = plain linear), ReLU after
  agg_init<<<cdiv(GH, 256), 256, 0, stream>>>(rp, degG, tmpG, GG);
  agg_edges<<<cdiv((long)NE1 * HD, 256), 256, 0, stream>>>(e_co, e_co + NE1, nrm1, rp, tmpG, NE1);
  gemm_wmma<HD,false,false,true,false><<<gblk(GG,HD),32*WPB,0,stream>>>(
      tmpG, coexW, HD, coexB, posA, GG, HD, nullptr, nullptr, nullptr, nullptr, 0.f);
  gemm_wmma<HD,false,false,true,false><<<gblk(GG,HD),32*WPB,0,stream>>>(
      rp, coexW, HD, coexB, posB, GG, HD, nullptr, nullptr, nullptr, nullptr, 0.f);
  // layer 1 (no ReLU)
  agg_init<<<cdiv(GH, 256), 256, 0, stream>>>(posA, degG, tmpG, GG);
  agg_edges<<<cdiv((long)NE1 * HD, 256), 256, 0, stream>>>(e_co, e_co + NE1, nrm1, posA, tmpG, NE1);
  gemm_wmma<HD,false,false,false,false><<<gblk(GG,HD),32*WPB,0,stream>>>(
      tmpG, coexW + HD*HD, HD, coexB + HD, posA2, GG, HD, nullptr, nullptr, nullptr, nullptr, 0.f);
  gemm_wmma<HD,false,false,false,false><<<gblk(GG,HD),32*WPB,0,stream>>>(
      posB, coexW + HD*HD, HD, coexB + HD, posB2, GG, HD, nullptr, nullptr, nullptr, nullptr, 0.f);

  // --- GO graph SGConv stack on pert embeddings
  deg_init<<<cdiv(PP, 256), 256, 0, stream>>>(degP, PP);
  deg_acc<<<cdiv(NE2, 256), 256, 0, stream>>>(e_go + NE2, w_go, degP, NE2);
  edge_norm<<<cdiv(NE2, 256), 256, 0, stream>>>(e_go, e_go + NE2, w_go, degP, nrm2, NE2);
  agg_init<<<cdiv(PH, 256), 256, 0, stream>>>(rpe, degP, tmpP, PP);
  agg_edges<<<cdiv((long)NE2 * HD, 256), 256, 0, stream>>>(e_go, e_go + NE2, nrm2, rpe, tmpP, NE2);
  gemm_wmma<HD,false,false,true,false><<<gblk(PP,HD),32*WPB,0,stream>>>(
      tmpP, goW, HD, goB, pge1, PP, HD, nullptr, nullptr, nullptr, nullptr, 0.f);
  agg_init<<<cdiv(PH, 256), 256, 0, stream>>>(pge1, degP, tmpP, PP);
  agg_edges<<<cdiv((long)NE2 * HD, 256), 256, 0, stream>>>(e_go, e_go + NE2, nrm2, pge1, tmpP, NE2);
  gemm_wmma<HD,false,false,false,false><<<gblk(PP,HD),32*WPB,0,stream>>>(
      tmpP, goW + HD*HD, HD, goB + HD, pge2, PP, HD, nullptr, nullptr, nullptr, nullptr, 0.f);

  // --- etv2 MLP on the two row-sets (BN over N via weighted stats 1 / 127)
  axpy02<<<cdiv(GH, 256), 256, 0, stream>>>(gnorm, posA2, inA, (int)GH);
  axpy02<<<cdiv(GH, 256), 256, 0, stream>>>(gnorm, posB2, inB, (int)GH);
  gemm_wmma<HD,false,false,false,true><<<gblk(GG,HD),32*WPB,0,stream>>>(
      inA, e_w1, HD, e_b1, hA, GG, HD, nullptr, nullptr, S(4), S(5), 1.f);
  gemm_wmma<HD,false,false,false,true><<<gblk(GG,HD),32*WPB,0,stream>>>(
      inB, e_w1, HD, e_b1, hB, GG, HD, nullptr, nullptr, S(4), S(5), 127.f);
  bn_finalize<<<1, 128, 0, stream>>>(S(4), S(5), (float)NTOT, e_g1, e_bb1, S(6), S(7), HD);
  gemm_wmma<HD,true,true,false,true><<<gblk(GG,HD),32*WPB,0,stream>>>(
      hA, e_w2, HD, e_b2, outA, GG, HD, S(6), S(7), S(8), S(9), 1.f);
  gemm_wmma<HD,true,true,false,true><<<gblk(GG,HD),32*WPB,0,stream>>>(
      hB, e_w2, HD, e_b2, outB, GG, HD, S(6), S(7), S(8), S(9), 127.f);
  bn_finalize<<<1, 128, 0, stream>>>(S(8), S(9), (float)NTOT, e_g2, e_bb2, S(10), S(11), HD);
  bn_apply<<<cdiv(GH, 256), 256, 0, stream>>>(outA, S(10), S(11), outA, GG, HD, 0);
  bn_apply<<<cdiv(GH, 256), 256, 0, stream>>>(outB, S(10), S(11), outB, GG, HD, 0);

  // --- pert_fuse MLP on gathered rows [B,H]
  gather_rows<<<cdiv(BH, 256), 256, 0, stream>>>(pge2, p_idx, fIn, BATCH);
  small_linear<<<cdiv(BH, 256), 256, 0, stream>>>(fIn, pf_w1, pf_b1, fh, BATCH, HD, HD);
  colstats<<<cdiv(BH, 256), 256, 0, stream>>>(fh, BATCH, HD, 1.f, S(12), S(13));
  bn_finalize<<<1, 128, 0, stream>>>(S(12), S(13), (float)BATCH, pf_g1, pf_bb1, S(14), S(15), HD);
  bn_apply<<<cdiv(BH, 256), 256, 0, stream>>>(fh, S(14), S(15), fh, BATCH, HD, 1);
  small_linear<<<cdiv(BH, 256), 256, 0, stream>>>(fh, pf_w2, pf_b2, fused, BATCH, HD, HD);
  colstats<<<cdiv(BH, 256), 256, 0, stream>>>(fused, BATCH, HD, 1.f, S(16), S(17));
  bn_finalize<<<1, 128, 0, stream>>>(S(16), S(17), (float)BATCH, pf_g2, pf_bb2, S(18), S(19), HD);
  bn_apply<<<cdiv(BH, 256), 256, 0, stream>>>(fused, S(18), S(19), fused, BATCH, HD, 0);

  // --- bn_pb stats (closed form from set + fused column sums)
  colstats<<<cdiv(GH, 256), 256, 0, stream>>>(outA, GG, HD, 1.f, S(20), S(21));
  colstats<<<cdiv(GH, 256), 256, 0, stream>>>(outB, GG, HD, 1.f, S(22), S(23));
  colstats<<<cdiv(BH, 256), 256, 0, stream>>>(fused, BATCH, HD, 1.f, S(24), S(25));
  pb_stats<<<1, 128, 0, stream>>>(S(20), S(21), S(22), S(23), S(24), S(25),
                                  fused, bnPg, bnPb, S(26), S(27));

  // --- recovery MLP at full N scale (the only genuinely-[N,*] compute)
  recov_l1_wmma<<<gblk(NTOT, 256), 32 * WPB, 0, stream>>>(
      outA, outB, fused, S(26), S(27), r_w1, r_b1, H1, S(28), S(29));
  bn_finalize<<<1, 256, 0, stream>>>(S(28), S(29), (float)NTOT, r_g1, r_bb1, S(30), S(31), 256);
  gemm_wmma<256,true,true,false,true><<<gblk(NTOT,HD),32*WPB,0,stream>>>(
      H1, r_w2, 256, r_b2, Oraw, NTOT, HD, S(30), S(31), S(32), S(33), 1.f);
  bn_finalize<<<1, 128, 0, stream>>>(S(32), S(33), (float)NTOT, r_g2, r_bb2, S(34), S(35), HD);

  // --- per-gene decode 1 (BN fused into the dot)
  decode1<<<cdiv((long)NTOT * 32, 256), 256, 0, stream>>>(Oraw, S(34), S(35), iw1, ib1, wmat);

  // --- cross-gene MLP
  cross_l1<<<cdiv((long)BATCH * HD * 32, 256), 256, 0, stream>>>(wmat, c_w1, c_b1, cg1);
  colstats<<<cdiv(BH, 256), 256, 0, stream>>>(cg1, BATCH, HD, 1.f, S(36), S(37));
  bn_finalize<<<1, 128, 0, stream>>>(S(36), S(37), (float)BATCH, c_g1, c_bb1, S(38), S(39), HD);
  bn_apply<<<cdiv(BH, 256), 256, 0, stream>>>(cg1, S(38), S(39), cgh, BATCH, HD, 1);
  small_linear<<<cdiv(BH, 256), 256, 0, stream>>>(cgh, c_w2, c_b2, cg, BATCH, HD, HD);
  colstats<<<cdiv(BH, 256), 256, 0, stream>>>(cg, BATCH, HD, 1.f, S(40), S(41));
  bn_finalize<<<1, 128, 0, stream>>>(S(40), S(41), (float)BATCH, c_g2, c_bb2, S(42), S(43), HD);
  bn_apply<<<cdiv(BH, 256), 256, 0, stream>>>(cg, S(42), S(43), cg, BATCH, HD, 0);

  // --- T[b,g] = cg @ iw2[:,1:]^T + ib2  (WMMA GEMM, ldw = H+1), then final sum
  gemm_wmma<HD,false,false,false,false><<<gblk(BATCH,GG),32*WPB,0,stream>>>(
      cg, iw2 + 1, HD + 1, ib2, Tbuf, BATCH, GG, nullptr, nullptr, nullptr, nullptr, 0.f);
  final_out<<<cdiv(NTOT, 256), 256, 0, stream>>>(Tbuf, wmat, iw2, x, out, NTOT);
}